// SpatioTemporalGAT_6622839571108
// MI455X (gfx1250) — compile-verified
//
#include <hip/hip_runtime.h>
#include <math.h>

typedef __attribute__((ext_vector_type(2))) float v2f;
typedef __attribute__((ext_vector_type(8))) float v8f;

#define NEG_SLOPE 0.2f

// ---------------------------------------------------------------------------
// Float atomic max via monotonic integer-bit mapping: lowers to native
// global_atomic_max_i32 / global_atomic_min_u32 (no CAS loop).
// ---------------------------------------------------------------------------
__device__ __forceinline__ void atomicMaxF(float* addr, float v) {
    if (v >= 0.0f) atomicMax((int*)addr, __float_as_int(v));
    else           atomicMin((unsigned int*)addr, __float_as_uint(v));
}

__device__ __forceinline__ float leaky(float v) {
    return v > 0.0f ? v : NEG_SLOPE * v;
}

// edge e in [0, EE): first E entries from edge_index, remainder self-loops
__device__ __forceinline__ void edge_sd(const int* __restrict__ ei, int E, int e,
                                        int& s, int& d) {
    if (e < E) { s = ei[e]; d = ei[E + e]; }
    else       { s = e - E; d = e - E; }
}

// ---------------------------------------------------------------------------
// 0) workspace init (graph-replay safe: runs every launch)
// ---------------------------------------------------------------------------
__global__ void k_init(float* m1, float* s1, float* out1,
                       float* m2, float* s2, float* out2, int n) {
    int i = blockIdx.x * blockDim.x + threadIdx.x;
    if (i < n * 64) out1[i] = 0.0f;
    if (i < n * 8)  { m1[i] = -INFINITY; s1[i] = 0.0f; }
    if (i < n * 2)  out2[i] = 0.0f;
    if (i < n)      { m2[i] = -INFINITY; s2[i] = 0.0f; }
}

// ---------------------------------------------------------------------------
// 1) h1[N,64] = x[N,3] @ W1[3,64] via V_WMMA_F32_16X16X4_F32 (K padded 3->4).
//    One wave per 16-node block; grid covers only full blocks -> no guards,
//    no EXEC manipulation around the WMMA. Pad lane zeroed via select.
// ---------------------------------------------------------------------------
__global__ void k_gemm1_wmma(const float* __restrict__ x,
                             const float* __restrict__ W1,
                             float* __restrict__ h1) {
    const int lane = threadIdx.x;             // 0..31
    const int base = blockIdx.x * 16;
    const int mrow = lane & 15;
    const bool hi  = lane >= 16;              // lane half selects K={0,1} / {2,pad}
    const int nrow = base + mrow;

    // A 16x4: unconditional loads at valid addresses; K=3 pad via cndmask
    float a0 = x[nrow * 3 + (hi ? 2 : 0)];
    float a1 = x[nrow * 3 + (hi ? 2 : 1)];    // value discarded for hi lanes
    v2f a;
    a.x = a0;
    a.y = hi ? 0.0f : a1;

    #pragma unroll
    for (int t = 0; t < 4; ++t) {
        const int col = t * 16 + mrow;
        float b0 = W1[(hi ? 128 : 0) + col];  // K=2 / K=0
        float b1 = W1[(hi ? 128 : 64) + col]; // discarded for hi (K=3 pad)
        v2f b;
        b.x = b0;
        b.y = hi ? 0.0f : b1;
        v8f c = {};
        c = __builtin_amdgcn_wmma_f32_16x16x4_f32(false, a, false, b,
                                                  (short)0, c, false, false);
        // D 16x16: lane half selects M base 0/8, N = lane&15
        float* dst = h1 + (size_t)(base + (hi ? 8 : 0)) * 64 + col;
        #pragma unroll
        for (int r = 0; r < 8; ++r)
            dst[r * 64] = c[r];               // unconditional, lane-coalesced
    }
}

// scalar tail for n % 16 rows (never launched for N=50000)
__global__ void k_gemm1_tail(const float* __restrict__ x,
                             const float* __restrict__ W1,
                             float* __restrict__ h1, int start, int n) {
    int i = start * 64 + blockIdx.x * blockDim.x + threadIdx.x;
    if (i >= n * 64) return;
    int row = i >> 6, col = i & 63;
    float acc = 0.0f;
    #pragma unroll
    for (int k = 0; k < 3; ++k) acc += x[row * 3 + k] * W1[k * 64 + col];
    h1[i] = acc;
}

// ---------------------------------------------------------------------------
// 2) per-(node,head) attention coefficients for layer 1
// ---------------------------------------------------------------------------
__global__ void k_att1(const float* __restrict__ h1,
                       const float* __restrict__ att_src,
                       const float* __restrict__ att_dst,
                       float* __restrict__ a_src, float* __restrict__ a_dst,
                       int n) {
    int i = blockIdx.x * blockDim.x + threadIdx.x;
    if (i >= n * 8) return;
    int node = i >> 3, h = i & 7;
    const float* hp = h1 + (size_t)node * 64 + h * 8;
    float ss = 0.0f, sd = 0.0f;
    #pragma unroll
    for (int c = 0; c < 8; ++c) {
        float v = hp[c];
        ss += v * att_src[h * 8 + c];
        sd += v * att_dst[h * 8 + c];
    }
    a_src[i] = ss;
    a_dst[i] = sd;
}

// ---------------------------------------------------------------------------
// 3-5) layer-1 edge passes: segment max, segment sum of exp, weighted agg.
//      Per-node 8-head vectors are 32B aligned -> b128 loads.
// ---------------------------------------------------------------------------
__global__ void k_edge1_max(const int* __restrict__ ei, int E, int EE,
                            const float* __restrict__ a_src,
                            const float* __restrict__ a_dst,
                            float* __restrict__ m1) {
    int e = blockIdx.x * blockDim.x + threadIdx.x;
    if (e >= EE) return;
    int s, d; edge_sd(ei, E, e, s, d);
    float4 s0 = ((const float4*)(a_src + (size_t)s * 8))[0];
    float4 s1v = ((const float4*)(a_src + (size_t)s * 8))[1];
    float4 d0 = ((const float4*)(a_dst + (size_t)d * 8))[0];
    float4 d1 = ((const float4*)(a_dst + (size_t)d * 8))[1];
    float vs[8] = {s0.x, s0.y, s0.z, s0.w, s1v.x, s1v.y, s1v.z, s1v.w};
    float vd[8] = {d0.x, d0.y, d0.z, d0.w, d1.x, d1.y, d1.z, d1.w};
    #pragma unroll
    for (int h = 0; h < 8; ++h)
        atomicMaxF(&m1[(size_t)d * 8 + h], leaky(vs[h] + vd[h]));
}

__global__ void k_edge1_sum(const int* __restrict__ ei, int E, int EE,
                            const float* __restrict__ a_src,
                            const float* __restrict__ a_dst,
                            const float* __restrict__ m1,
                            float* __restrict__ s1) {
    int e = blockIdx.x * blockDim.x + threadIdx.x;
    if (e >= EE) return;
    int s, d; edge_sd(ei, E, e, s, d);
    float4 s0 = ((const float4*)(a_src + (size_t)s * 8))[0];
    float4 s1v = ((const float4*)(a_src + (size_t)s * 8))[1];
    float4 d0 = ((const float4*)(a_dst + (size_t)d * 8))[0];
    float4 d1 = ((const float4*)(a_dst + (size_t)d * 8))[1];
    float4 m0 = ((const float4*)(m1 + (size_t)d * 8))[0];
    float4 m1v = ((const float4*)(m1 + (size_t)d * 8))[1];
    float vs[8] = {s0.x, s0.y, s0.z, s0.w, s1v.x, s1v.y, s1v.z, s1v.w};
    float vd[8] = {d0.x, d0.y, d0.z, d0.w, d1.x, d1.y, d1.z, d1.w};
    float vm[8] = {m0.x, m0.y, m0.z, m0.w, m1v.x, m1v.y, m1v.z, m1v.w};
    #pragma unroll
    for (int h = 0; h < 8; ++h) {
        float ex = __expf(leaky(vs[h] + vd[h]) - vm[h]);
        atomicAdd(&s1[(size_t)d * 8 + h], ex);
    }
}

// one thread per (edge, channel); consecutive threads cover the 64 channels
// of one edge -> coalesced h1 gather and coalesced atomic targets
__global__ void k_edge1_agg(const int* __restrict__ ei, int E, int EE,
                            const float* __restrict__ a_src,
                            const float* __restrict__ a_dst,
                            const float* __restrict__ m1,
                            const float* __restrict__ s1,
                            const float* __restrict__ h1,
                            float* __restrict__ out1) {
    unsigned int t = blockIdx.x * blockDim.x + threadIdx.x;
    unsigned int total = (unsigned int)EE * 64u;
    if (t >= total) return;
    int e = (int)(t >> 6);
    int c = (int)(t & 63u);
    int h = c >> 3;
    int s, d; edge_sd(ei, E, e, s, d);
    float v     = leaky(a_src[(size_t)s * 8 + h] + a_dst[(size_t)d * 8 + h]);
    float alpha = __expf(v - m1[(size_t)d * 8 + h]) / s1[(size_t)d * 8 + h];
    atomicAdd(&out1[(size_t)d * 64 + c], alpha * h1[(size_t)s * 64 + c]);
}

// ---------------------------------------------------------------------------
// 6) act1 = ELU(out1 + b1), in place
// ---------------------------------------------------------------------------
__global__ void k_elu(float* __restrict__ out1, const float* __restrict__ b1,
                      int n) {
    int i = blockIdx.x * blockDim.x + threadIdx.x;
    if (i >= n * 64) return;
    float v = out1[i] + b1[i & 63];
    out1[i] = v > 0.0f ? v : (__expf(v) - 1.0f);
}

// ---------------------------------------------------------------------------
// 7) h2[N,2] = act1[N,64] @ W2[64,2] via 16 accumulating f32 WMMA K-steps.
//    Full blocks only; B columns >=2 zeroed by select, not branch.
// ---------------------------------------------------------------------------
__global__ void k_gemm2_wmma(const float* __restrict__ act,
                             const float* __restrict__ W2,
                             float* __restrict__ h2) {
    const int lane = threadIdx.x;
    const int base = blockIdx.x * 16;
    const int mrow = lane & 15;
    const bool hi  = lane >= 16;
    const int nrow = base + mrow;
    const int mcol = mrow & 1;                 // clamped valid column
    const bool act_col = mrow < 2;

    v8f c = {};
    #pragma unroll
    for (int k = 0; k < 16; ++k) {
        const int k0 = 4 * k + (hi ? 2 : 0);
        v2f a, b;
        a.x = act[(size_t)nrow * 64 + k0];
        a.y = act[(size_t)nrow * 64 + k0 + 1];
        float b0 = W2[k0 * 2 + mcol];          // unconditional valid loads
        float b1 = W2[(k0 + 1) * 2 + mcol];
        b.x = act_col ? b0 : 0.0f;             // cndmask, no branch
        b.y = act_col ? b1 : 0.0f;
        c = __builtin_amdgcn_wmma_f32_16x16x4_f32(false, a, false, b,
                                                  (short)0, c, false, false);
    }
    if (act_col) {
        float* dst = h2 + (size_t)(base + (hi ? 8 : 0)) * 2 + mcol;
        #pragma unroll
        for (int r = 0; r < 8; ++r)
            dst[r * 2] = c[r];
    }
}

// scalar tail for n % 16 rows (never launched for N=50000)
__global__ void k_gemm2_tail(const float* __restrict__ act,
                             const float* __restrict__ W2,
                             float* __restrict__ h2, int start, int n) {
    int i = start * 2 + blockIdx.x * blockDim.x + threadIdx.x;
    if (i >= n * 2) return;
    int row = i >> 1, col = i & 1;
    float acc = 0.0f;
    for (int k = 0; k < 64; ++k) acc += act[(size_t)row * 64 + k] * W2[k * 2 + col];
    h2[i] = acc;
}

// ---------------------------------------------------------------------------
// 8) layer-2 attention coefficients (heads=1, ch=2)
// ---------------------------------------------------------------------------
__global__ void k_att2(const float* __restrict__ h2,
                       const float* __restrict__ att_src,
                       const float* __restrict__ att_dst,
                       float* __restrict__ a_src, float* __restrict__ a_dst,
                       int n) {
    int i = blockIdx.x * blockDim.x + threadIdx.x;
    if (i >= n) return;
    float v0 = h2[i * 2 + 0], v1 = h2[i * 2 + 1];
    a_src[i] = v0 * att_src[0] + v1 * att_src[1];
    a_dst[i] = v0 * att_dst[0] + v1 * att_dst[1];
}

// ---------------------------------------------------------------------------
// 9-11) layer-2 edge passes
// ---------------------------------------------------------------------------
__global__ void k_edge2_max(const int* __restrict__ ei, int E, int EE,
                            const float* __restrict__ a_src,
                            const float* __restrict__ a_dst,
                            float* __restrict__ m2) {
    int e = blockIdx.x * blockDim.x + threadIdx.x;
    if (e >= EE) return;
    int s, d; edge_sd(ei, E, e, s, d);
    atomicMaxF(&m2[d], leaky(a_src[s] + a_dst[d]));
}

__global__ void k_edge2_sum(const int* __restrict__ ei, int E, int EE,
                            const float* __restrict__ a_src,
                            const float* __restrict__ a_dst,
                            const float* __restrict__ m2,
                            float* __restrict__ s2) {
    int e = blockIdx.x * blockDim.x + threadIdx.x;
    if (e >= EE) return;
    int s, d; edge_sd(ei, E, e, s, d);
    float v = leaky(a_src[s] + a_dst[d]);
    atomicAdd(&s2[d], __expf(v - m2[d]));
}

__global__ void k_edge2_agg(const int* __restrict__ ei, int E, int EE,
                            const float* __restrict__ a_src,
                            const float* __restrict__ a_dst,
                            const float* __restrict__ m2,
                            const float* __restrict__ s2,
                            const float* __restrict__ h2,
                            float* __restrict__ out2) {
    unsigned int t = blockIdx.x * blockDim.x + threadIdx.x;
    unsigned int total = (unsigned int)EE * 2u;
    if (t >= total) return;
    int e = (int)(t >> 1);
    int c = (int)(t & 1u);
    int s, d; edge_sd(ei, E, e, s, d);
    float v     = leaky(a_src[s] + a_dst[d]);
    float alpha = __expf(v - m2[d]) / s2[d];
    atomicAdd(&out2[(size_t)d * 2 + c], alpha * h2[(size_t)s * 2 + c]);
}

// ---------------------------------------------------------------------------
// 12) out = log_softmax(out2 + b2) over the 2 classes
// ---------------------------------------------------------------------------
__global__ void k_final(const float* __restrict__ out2,
                        const float* __restrict__ b2,
                        float* __restrict__ out, int n) {
    int i = blockIdx.x * blockDim.x + threadIdx.x;
    if (i >= n) return;
    float z0 = out2[i * 2 + 0] + b2[0];
    float z1 = out2[i * 2 + 1] + b2[1];
    float mx = fmaxf(z0, z1);
    float ls = mx + __logf(__expf(z0 - mx) + __expf(z1 - mx));
    out[i * 2 + 0] = z0 - ls;
    out[i * 2 + 1] = z1 - ls;
}

// ---------------------------------------------------------------------------
extern "C" void kernel_launch(void* const* d_in, const int* in_sizes, int n_in,
                              void* d_out, int out_size, void* d_ws, size_t ws_size,
                              hipStream_t stream) {
    const float* x        = (const float*)d_in[0];
    const int*   ei       = (const int*)d_in[1];   // [2, E]
    const float* W1       = (const float*)d_in[2]; // [3, 64]
    const float* att_src1 = (const float*)d_in[3]; // [8, 8]
    const float* att_dst1 = (const float*)d_in[4]; // [8, 8]
    const float* b1       = (const float*)d_in[5]; // [64]
    const float* W2       = (const float*)d_in[6]; // [64, 2]
    const float* att_src2 = (const float*)d_in[7]; // [1, 2]
    const float* att_dst2 = (const float*)d_in[8]; // [1, 2]
    const float* b2       = (const float*)d_in[9]; // [2]
    float* out = (float*)d_out;

    const int n  = in_sizes[0] / 3;   // 50000
    const int E  = in_sizes[1] / 2;   // 1600000
    const int EE = E + n;             // + self loops

    // workspace layout (floats): total n*168 (~33.6 MB)
    float* ws     = (float*)d_ws;
    float* h1     = ws;               // n*64
    float* a_src1 = h1     + n * 64;  // n*8
    float* a_dst1 = a_src1 + n * 8;   // n*8
    float* m1     = a_dst1 + n * 8;   // n*8
    float* s1     = m1     + n * 8;   // n*8
    float* out1   = s1     + n * 8;   // n*64 (becomes act1 after ELU)
    float* h2     = out1   + n * 64;  // n*2
    float* a_src2 = h2     + n * 2;   // n
    float* a_dst2 = a_src2 + n;       // n
    float* m2     = a_dst2 + n;       // n
    float* s2     = m2     + n;       // n
    float* out2   = s2     + n;       // n*2

    const int B = 256;
    const int nfull = n / 16;         // full 16-row WMMA blocks
    const int rem   = n - nfull * 16; // tail rows (0 for N=50000)

    k_init<<<(n * 64 + B - 1) / B, B, 0, stream>>>(m1, s1, out1, m2, s2, out2, n);

    if (nfull > 0) k_gemm1_wmma<<<nfull, 32, 0, stream>>>(x, W1, h1);
    if (rem > 0)
        k_gemm1_tail<<<(rem * 64 + B - 1) / B, B, 0, stream>>>(x, W1, h1,
                                                               nfull * 16, n);
    k_att1<<<(n * 8 + B - 1) / B, B, 0, stream>>>(h1, att_src1, att_dst1,
                                                  a_src1, a_dst1, n);

    k_edge1_max<<<(EE + B - 1) / B, B, 0, stream>>>(ei, E, EE, a_src1, a_dst1, m1);
    k_edge1_sum<<<(EE + B - 1) / B, B, 0, stream>>>(ei, E, EE, a_src1, a_dst1, m1, s1);
    {
        unsigned int total = (unsigned int)EE * 64u;
        k_edge1_agg<<<(total + B - 1) / B, B, 0, stream>>>(ei, E, EE, a_src1, a_dst1,
                                                           m1, s1, h1, out1);
    }

    k_elu<<<(n * 64 + B - 1) / B, B, 0, stream>>>(out1, b1, n);

    if (nfull > 0) k_gemm2_wmma<<<nfull, 32, 0, stream>>>(out1, W2, h2);
    if (rem > 0)
        k_gemm2_tail<<<(rem * 2 + B - 1) / B, B, 0, stream>>>(out1, W2, h2,
                                                              nfull * 16, n);
    k_att2<<<(n + B - 1) / B, B, 0, stream>>>(h2, att_src2, att_dst2,
                                              a_src2, a_dst2, n);

    k_edge2_max<<<(EE + B - 1) / B, B, 0, stream>>>(ei, E, EE, a_src2, a_dst2, m2);
    k_edge2_sum<<<(EE + B - 1) / B, B, 0, stream>>>(ei, E, EE, a_src2, a_dst2, m2, s2);
    {
        unsigned int total = (unsigned int)EE * 2u;
        k_edge2_agg<<<(total + B - 1) / B, B, 0, stream>>>(ei, E, EE, a_src2, a_dst2,
                                                           m2, s2, h2, out2);
    }

    k_final<<<(n + B - 1) / B, B, 0, stream>>>(out2, b2, out, n);
}